// DHN_84696755077556
// MI455X (gfx1250) — compile-verified
//
#include <hip/hip_runtime.h>

typedef __attribute__((ext_vector_type(2))) float v2f;
typedef __attribute__((ext_vector_type(8))) float v8f;

#define NCOL 192   // concat feature width (3 kernels x 64)
#define DK   64

// ---------------------------------------------------------------------------
// Pack three [K x 64] weight mats + biases into one [K x 192] + [192]
// ---------------------------------------------------------------------------
__global__ __launch_bounds__(256) void pack_k(
    const float* __restrict__ W0, const float* __restrict__ W1,
    const float* __restrict__ W2, const float* __restrict__ b0,
    const float* __restrict__ b1, const float* __restrict__ b2,
    float* __restrict__ Wc, float* __restrict__ bc, int Kdim)
{
    int idx = blockIdx.x * blockDim.x + threadIdx.x;
    if (idx >= Kdim * NCOL) return;
    int k = idx / NCOL, c = idx % NCOL;
    int j = c >> 6, cc = c & 63;
    const float* W = (j == 0) ? W0 : ((j == 1) ? W1 : W2);
    Wc[idx] = W[k * DK + cc];
    if (k == 0) {
        const float* bb = (j == 0) ? b0 : ((j == 1) ? b1 : b2);
        bc[c] = bb[cc];
    }
}

// ---------------------------------------------------------------------------
// H[M x 192] = (RELU_IN ? relu(A) : A)[M x K] @ W[K x 192] + bias
// One wave per 16x16 output tile, fp32 WMMA (16x16x4).
// ---------------------------------------------------------------------------
template<bool RELU_IN>
__global__ __launch_bounds__(256) void gemm_wmma_k(
    const float* __restrict__ A, const float* __restrict__ W,
    const float* __restrict__ bias, float* __restrict__ H, int M, int K)
{
    int gwave = (blockIdx.x * blockDim.x + threadIdx.x) >> 5;
    int lane  = threadIdx.x & 31;
    int mtile = gwave / (NCOL / 16);
    int ntile = gwave % (NCOL / 16);
    if (mtile * 16 >= M) return;          // wave-uniform: EXEC stays all-1s

    int m  = lane & 15;
    int kq = (lane >> 4) << 1;            // 0 or 2
    int n0 = ntile * 16;
    const float* arow = A + (size_t)(mtile * 16 + m) * K;
    const float* wcol = W + n0 + m;

    v8f c = {};
    for (int k = 0; k < K; k += 4) {
        float a0 = arow[k + kq];
        float a1 = arow[k + kq + 1];
        if (RELU_IN) { a0 = fmaxf(a0, 0.f); a1 = fmaxf(a1, 0.f); }
        v2f av, bv;
        av.x = a0;
        av.y = a1;
        bv.x = wcol[(size_t)(k + kq) * NCOL];
        bv.y = wcol[(size_t)(k + kq + 1) * NCOL];
        c = __builtin_amdgcn_wmma_f32_16x16x4_f32(
                false, av, false, bv, (short)0, c, false, false);
    }

    int col   = n0 + m;
    int rbase = mtile * 16 + ((lane >> 4) << 3);
    float bb  = bias[col];
#pragma unroll
    for (int v = 0; v < 8; ++v)
        H[(size_t)(rbase + v) * NCOL + col] = c[v] + bb;
}

// ---------------------------------------------------------------------------
// CSR construction: per-destination edge lists for all 6 maps.
// ---------------------------------------------------------------------------
__device__ __forceinline__ const int* pick_map(
    const int* m0, const int* m1, const int* m2,
    const int* m3, const int* m4, const int* m5, int y)
{
    const int* p = m0;
    if (y == 1) p = m1; else if (y == 2) p = m2;
    else if (y == 3) p = m3; else if (y == 4) p = m4;
    else if (y == 5) p = m5;
    return p;
}

__global__ __launch_bounds__(256) void csr_count_k(
    const int* m0, const int* m1, const int* m2,
    const int* m3, const int* m4, const int* m5,
    int* __restrict__ counts, int N, int E)
{
    const int* map = pick_map(m0, m1, m2, m3, m4, m5, blockIdx.y);
    int* cnt = counts + (size_t)blockIdx.y * N;
    int e = blockIdx.x * blockDim.x + threadIdx.x;
    if (e >= E) return;
    atomicAdd(&cnt[map[e]], 1);           // map row 0 = destinations (roots)
}

// One block per map (6 blocks), 1024-thread block-wide exclusive scan.
__global__ __launch_bounds__(1024) void csr_scan_k(
    const int* __restrict__ counts, int* __restrict__ rowptr,
    int* __restrict__ cursor, int N)
{
    const int* cnt = counts + (size_t)blockIdx.x * N;
    int* rp  = rowptr + (size_t)blockIdx.x * (N + 1);
    int* cur = cursor + (size_t)blockIdx.x * N;
    __shared__ int part[1024];
    int t = threadIdx.x;
    int chunk = (N + 1023) >> 10;
    int beg = t * chunk;
    int end = min(beg + chunk, N);
    int s = 0;
    for (int i = beg; i < end; ++i) s += cnt[i];
    part[t] = s;
    __syncthreads();
    // Hillis-Steele inclusive scan over the 1024 partials
    for (int d = 1; d < 1024; d <<= 1) {
        int add = (t >= d) ? part[t - d] : 0;
        __syncthreads();
        part[t] += add;
        __syncthreads();
    }
    int pre = (t == 0) ? 0 : part[t - 1];
    for (int i = beg; i < end; ++i) {
        rp[i] = pre; cur[i] = pre;
        pre += cnt[i];
    }
    if (t == 1023) rp[N] = part[1023];
}

__global__ __launch_bounds__(256) void csr_fill_k(
    const int* m0, const int* m1, const int* m2,
    const int* m3, const int* m4, const int* m5,
    int* __restrict__ cursor, int* __restrict__ eidx, int N, int E)
{
    const int* map = pick_map(m0, m1, m2, m3, m4, m5, blockIdx.y);
    int* cur = cursor + (size_t)blockIdx.y * N;
    int* ei  = eidx   + (size_t)blockIdx.y * E;
    int e = blockIdx.x * blockDim.x + threadIdx.x;
    if (e >= E) return;
    int dst = map[e];
    int src = map[E + e];
    int slot = atomicAdd(&cur[dst], 1);
    ei[slot] = src;                        // store gather source per dst-slot
}

// ---------------------------------------------------------------------------
// Atomic-free segment sum: F[n, kofs+j] = sum over row-n edges of H[src, kofs+j]
// One wave per (node, 64-col group): lane handles cols {lane, lane+32}.
// Zero-degree rows write zeros (replaces memset of F).
// ---------------------------------------------------------------------------
__global__ __launch_bounds__(256) void spmm_k(
    const float* __restrict__ H, float* __restrict__ F,
    const int* __restrict__ rowptr, const int* __restrict__ eidx,
    int csrbase, int N, int E)
{
    int kk = blockIdx.y;                   // 0..2 (which hom-kernel)
    const int* rp = rowptr + (size_t)(csrbase + kk) * (N + 1);
    const int* ei = eidx   + (size_t)(csrbase + kk) * E;
    int kofs = kk * DK;
    int gwave = (blockIdx.x * blockDim.x + threadIdx.x) >> 5;
    int lane  = threadIdx.x & 31;
    if (gwave >= N) return;
    int r0 = rp[gwave], r1 = rp[gwave + 1];
    float acc0 = 0.f, acc1 = 0.f;
    for (int j = r0; j < r1; ++j) {
        int src = ei[j];
        const float* hp = H + (size_t)src * NCOL + kofs;
        acc0 += hp[lane];
        acc1 += hp[lane + 32];
    }
    float* fp = F + (size_t)gwave * NCOL + kofs;
    fp[lane]      = acc0;
    fp[lane + 32] = acc1;
}

// ---------------------------------------------------------------------------
// pooled[b, c] += relu(F[n, c]); batch_idx sorted -> run-length atomic flush.
// ---------------------------------------------------------------------------
__global__ __launch_bounds__(256) void pool_k(
    const float* __restrict__ F, const int* __restrict__ bidx,
    float* __restrict__ pooled, int M)
{
    int nblocks = (M + 7) >> 3;
    int idx = blockIdx.x * blockDim.x + threadIdx.x;
    if (idx >= nblocks * NCOL) return;
    int c  = idx % NCOL;
    int nb = idx / NCOL;
    int n0 = nb * 8;
    int n1 = min(n0 + 8, M);
    float acc = 0.f;
    int cur = bidx[n0];
    for (int n = n0; n < n1; ++n) {
        int b = bidx[n];
        if (b != cur) {
            unsafeAtomicAdd(&pooled[cur * NCOL + c], acc);
            acc = 0.f; cur = b;
        }
        acc += fmaxf(F[(size_t)n * NCOL + c], 0.f);
    }
    unsafeAtomicAdd(&pooled[cur * NCOL + c], acc);
}

__global__ __launch_bounds__(256) void mlp1_k(
    const float* __restrict__ pooled, const float* __restrict__ A1,
    const float* __restrict__ ba1, float* __restrict__ hidden)
{
    int idx = blockIdx.x * blockDim.x + threadIdx.x;
    if (idx >= 64 * 256) return;
    int b = idx >> 8, o = idx & 255;
    float acc = ba1[o];
    for (int c = 0; c < NCOL; ++c)
        acc += pooled[b * NCOL + c] * A1[c * 256 + o];
    hidden[idx] = fmaxf(acc, 0.f);
}

__global__ __launch_bounds__(256) void mlp2_k(
    const float* __restrict__ hidden, const float* __restrict__ A2,
    const float* __restrict__ ba2, float* __restrict__ out)
{
    int idx = blockIdx.x * blockDim.x + threadIdx.x;
    if (idx >= 64 * 10) return;
    int b = idx / 10, o = idx % 10;
    float acc = ba2[o];
    for (int h = 0; h < 256; ++h)
        acc += hidden[b * 256 + h] * A2[h * 10 + o];
    out[idx] = acc;
}

// ---------------------------------------------------------------------------
extern "C" void kernel_launch(void* const* d_in, const int* in_sizes, int n_in,
                              void* d_out, int out_size, void* d_ws, size_t ws_size,
                              hipStream_t stream)
{
    const float* x     = (const float*)d_in[0];
    const int*   map00 = (const int*)d_in[1];
    const int*   map01 = (const int*)d_in[2];
    const int*   map02 = (const int*)d_in[3];
    const int*   map10 = (const int*)d_in[4];
    const int*   map11 = (const int*)d_in[5];
    const int*   map12 = (const int*)d_in[6];
    const int*   bidx  = (const int*)d_in[7];
    // d_in[8] = batch_size (device scalar); B fixed at 64 by the reference.
    const float* W00 = (const float*)d_in[9];
    const float* b00 = (const float*)d_in[10];
    const float* W01 = (const float*)d_in[11];
    const float* b01 = (const float*)d_in[12];
    const float* W02 = (const float*)d_in[13];
    const float* b02 = (const float*)d_in[14];
    const float* W10 = (const float*)d_in[15];
    const float* b10 = (const float*)d_in[16];
    const float* W11 = (const float*)d_in[17];
    const float* b11 = (const float*)d_in[18];
    const float* W12 = (const float*)d_in[19];
    const float* b12 = (const float*)d_in[20];
    const float* A1  = (const float*)d_in[21];
    const float* ba1 = (const float*)d_in[22];
    const float* A2  = (const float*)d_in[23];
    const float* ba2 = (const float*)d_in[24];
    float* out = (float*)d_out;

    const int D_IN = 128;
    const int N = in_sizes[0] / D_IN;   // 50000
    const int E = in_sizes[1] / 2;      // 800000

    // ---- workspace carve-up ----
    float* ws = (float*)d_ws;
    size_t off = 0;
    float* h0     = ws + off; off += (size_t)N * NCOL;
    float* f0     = ws + off; off += (size_t)N * NCOL;
    float* h1     = ws + off; off += (size_t)N * NCOL;
    float* f1     = ws + off; off += (size_t)N * NCOL;
    float* Wc0    = ws + off; off += (size_t)D_IN * NCOL;
    float* bc0    = ws + off; off += NCOL;
    float* Wc1    = ws + off; off += (size_t)NCOL * NCOL;
    float* bc1    = ws + off; off += NCOL;
    float* pooled = ws + off; off += (size_t)64 * NCOL;
    float* hidden = ws + off; off += (size_t)64 * 256;
    int* iws = (int*)(ws + off);
    size_t ioff = 0;
    int* counts = iws + ioff; ioff += (size_t)6 * N;
    int* rowptr = iws + ioff; ioff += (size_t)6 * (N + 1);
    int* cursor = iws + ioff; ioff += (size_t)6 * N;
    int* eidx   = iws + ioff; ioff += (size_t)6 * E;

    // ---- zero the accumulators ----
    hipMemsetAsync(counts, 0, (size_t)6 * N * sizeof(int), stream);
    hipMemsetAsync(pooled, 0, (size_t)64 * NCOL * sizeof(float), stream);

    // ---- build 6 CSRs (count -> scan -> fill) ----
    int ebx = (E + 255) / 256;
    csr_count_k<<<dim3(ebx, 6), 256, 0, stream>>>(
        map00, map01, map02, map10, map11, map12, counts, N, E);
    csr_scan_k<<<6, 1024, 0, stream>>>(counts, rowptr, cursor, N);
    csr_fill_k<<<dim3(ebx, 6), 256, 0, stream>>>(
        map00, map01, map02, map10, map11, map12, cursor, eidx, N, E);

    // ---- pack weights ----
    pack_k<<<(D_IN * NCOL + 255) / 256, 256, 0, stream>>>(
        W00, W01, W02, b00, b01, b02, Wc0, bc0, D_IN);
    pack_k<<<(NCOL * NCOL + 255) / 256, 256, 0, stream>>>(
        W10, W11, W12, b10, b11, b12, Wc1, bc1, NCOL);

    // ---- layer 0: h0 = x @ Wc0 + bc0 ; f0 = segsum(h0) ----
    int mtiles = (N + 15) / 16;
    int gblocks = (int)(((long long)mtiles * (NCOL / 16) * 32 + 255) / 256);
    gemm_wmma_k<false><<<gblocks, 256, 0, stream>>>(x, Wc0, bc0, h0, N, D_IN);
    int sblocks = (int)(((long long)N * 32 + 255) / 256);
    spmm_k<<<dim3(sblocks, 3), 256, 0, stream>>>(h0, f0, rowptr, eidx, 0, N, E);

    // ---- layer 1: h1 = relu(f0) @ Wc1 + bc1 ; f1 = segsum(h1) ----
    gemm_wmma_k<true><<<gblocks, 256, 0, stream>>>(f0, Wc1, bc1, h1, N, NCOL);
    spmm_k<<<dim3(sblocks, 3), 256, 0, stream>>>(h1, f1, rowptr, eidx, 3, N, E);

    // ---- pooling + MLP ----
    int pthreads = ((N + 7) / 8) * NCOL;
    pool_k<<<(pthreads + 255) / 256, 256, 0, stream>>>(f1, bidx, pooled, N);
    mlp1_k<<<(64 * 256 + 255) / 256, 256, 0, stream>>>(pooled, A1, ba1, hidden);
    mlp2_k<<<(64 * 10 + 255) / 256, 256, 0, stream>>>(hidden, A2, ba2, out);
}